// VQPromptBlock_86964497809966
// MI455X (gfx1250) — compile-verified
//
#include <hip/hip_runtime.h>

// ---------------------------------------------------------------------------
// VQ-VAE decode block for MI455X (gfx1250, wave32). All GEMM-like work runs on
// V_WMMA_F32_16X16X4_F32 to keep fp32 semantics (argmin tie sensitivity) while
// getting matrix-pipe throughput (2048 FLOP/instr vs 64 for v_fma_f32).
// Round 2: 3x3-conv weights pre-packed into K-pair-contiguous layout so the
// dominant GEMM loop is 1 ds_load_b64 + 1 global_load_b64 per WMMA (was
// 1 ds_b64 + 2 scalar global_b32).
// ---------------------------------------------------------------------------

typedef float v2f __attribute__((ext_vector_type(2)));
typedef float v8f __attribute__((ext_vector_type(8)));

#define B_    16
#define C_    256
#define H_    64
#define W_    64
#define OUTC  3
#define NT    512
#define CB    64

#define HW     (H_*W_)        // 4096
#define NPOS   (B_*HW)        // 65536 spatial positions
#define NTILE  (NPOS/16)      // 4096 16-row spatial tiles
#define CHW    (C_*HW)        // 1048576 per-batch stride of input/q

// workspace layout (float units)
#define Q_OFF      0u          // q tensor [B,C,H,W]           16,777,216 f
#define H_OFF      16777216u   // bottleneck h [B,CB,H,W]       4,194,304 f
#define EPACK_OFF  20971520u   // embed pair-packed [128][512]x2  131,072 f
#define EN2_OFF    21102592u   // ||e_n||^2                           512 f
#define CNT_OFF    21103104u   // uint counts [B][512]              8,192 u32
#define WDIFF_OFF  21111296u   // per-block diff partials           4,096 f
#define WPACK_OFF  21115392u   // packed w1 [3][9][128][64]x2     442,368 f
// total ~86.2 MB of d_ws

#define WP_PER_BLK (9*128*CB)  // 73728 float2 per res block
#define WP_PER_TAP (128*CB)    // 8192 float2 per tap

// output layout: quant [B,3,H,W] (196608) | diff (1) | wordfreq [B,512] (8192)
#define DIFF_POS   196608
#define WF_POS     196609

static __device__ __forceinline__ v8f wmma4(v2f a, v2f b, v8f c) {
  // V_WMMA_F32_16X16X4_F32 : D = A(16x4) * B(4x16) + C (fp32 throughout)
  return __builtin_amdgcn_wmma_f32_16x16x4_f32(false, a, false, b, (short)0, c,
                                               false, false);
}

// Decode spatial tile id -> (batch, row, col-base). 256 tiles per batch.
#define DECODE_TILE(s, b, hh, w0)                        \
  const int b  = (s) >> 8;                               \
  const int hh = ((s) & 255) >> 2;                       \
  const int w0 = ((s) & 3) << 4;

// ---------------------------------------------------------------------------
// K0: repack codebook into channel-pair-major form (one b64 load per WMMA B
// fragment), precompute per-code squared norms, zero histogram counters.
// ---------------------------------------------------------------------------
__global__ __launch_bounds__(256) void k_prep(const float* __restrict__ embed,
                                              float* __restrict__ epack,
                                              float* __restrict__ en2,
                                              unsigned* __restrict__ counts) {
  const int gid = blockIdx.x * 256 + threadIdx.x;   // grid covers 65536
  if (gid < 128 * NT) {
    const int cc = gid >> 9, n = gid & (NT - 1);
    epack[gid * 2 + 0] = embed[(2 * cc) * NT + n];
    epack[gid * 2 + 1] = embed[(2 * cc + 1) * NT + n];
  }
  if (gid < NT) {
    float s = 0.f;
    for (int c = 0; c < C_; ++c) { float v = embed[c * NT + gid]; s += v * v; }
    en2[gid] = s;
  }
  if (gid < B_ * NT) counts[gid] = 0u;
}

// ---------------------------------------------------------------------------
// K0b: repack res_w1 [3][64][256][3][3] -> wpack [i][tap][cc][oc] float2 with
// value = {w1[oc][2cc][dh][dw], w1[oc][2cc+1][dh][dw]}. Makes the 3x3-conv
// B fragment a single lane-coalesced global_load_b64 (consecutive oc across
// lanes = 128B bursts; K pair contiguous within a lane).
// ---------------------------------------------------------------------------
__global__ __launch_bounds__(256) void k_packw(const float* __restrict__ rw1,
                                               float* __restrict__ wpack) {
  const int t = blockIdx.x * 256 + threadIdx.x;     // grid covers 3*73728
  if (t >= 3 * WP_PER_BLK) return;
  const int i   = t / WP_PER_BLK;
  const int r   = t % WP_PER_BLK;
  const int tap = r / WP_PER_TAP;
  const int r2  = r % WP_PER_TAP;
  const int cc  = r2 >> 6;          // 0..127 (channel pair)
  const int oc  = r2 & 63;          // 0..63
  const int dh = tap / 3, dw = tap % 3;
  const float* src = rw1 + (size_t)i * CB * C_ * 9
                         + ((size_t)oc * C_ + 2 * cc) * 9 + dh * 3 + dw;
  wpack[t * 2 + 0] = src[0];
  wpack[t * 2 + 1] = src[9];        // next channel, same tap
}

// ---------------------------------------------------------------------------
// K1: VQ. Each block = 4 waves, one 16-row spatial tile. dist = ||e||^2 - 2 f.e
// (||f||^2 constant per row -> irrelevant for argmin). Each wave sweeps 8 code
// tiles of 16 via 64 fp32 WMMAs each (K=256). First-index tie-breaking matches
// jnp.argmin. Then writes quantized q, integer histogram, diff partial.
// ---------------------------------------------------------------------------
__global__ __launch_bounds__(128) void k_vq(const float* __restrict__ in,
                                            const float* __restrict__ embed,
                                            const float* __restrict__ epack,
                                            const float* __restrict__ en2,
                                            float* __restrict__ q,
                                            unsigned* __restrict__ counts,
                                            float* __restrict__ wsdiff) {
  __shared__ float As[4096];        // A tile pair-packed: [(cc*16+m)*2]
  __shared__ float rv[4][16][16];   // per-wave per-row candidates
  __shared__ int   ri[4][16][16];
  __shared__ int   idxs[16];
  __shared__ float red[128];

  const int s = blockIdx.x;
  DECODE_TILE(s, b, hh, w0)
  const int tid = threadIdx.x;
  const long gbase0 = (long)b * CHW + hh * W_ + w0;

  // stage x[c][m] (m = 16 consecutive w) as channel pairs for b64 A fragments
  for (int t = tid; t < 2048; t += 128) {
    const int cc = t >> 4, m = t & 15;
    As[t * 2 + 0] = in[gbase0 + (2 * cc) * HW + m];
    As[t * 2 + 1] = in[gbase0 + (2 * cc + 1) * HW + m];
  }
  __syncthreads();

  const int wave = tid >> 5, lane = tid & 31, g = lane >> 4, mc = lane & 15;
  float bestv[8];
  int   besti[8];
#pragma unroll
  for (int r = 0; r < 8; ++r) { bestv[r] = 3.4e38f; besti[r] = 0; }

  for (int tt = 0; tt < 8; ++tt) {
    const int n0 = (wave * 8 + tt) << 4;
    v8f acc = {0.f, 0.f, 0.f, 0.f, 0.f, 0.f, 0.f, 0.f};
#pragma unroll 8
    for (int k0 = 0; k0 < C_; k0 += 4) {
      const int cc = (k0 >> 1) + g;                       // channels k0+2g, +1
      v2f a  = *(const v2f*)(As + (cc * 16 + mc) * 2);    // ds_load_b64
      v2f bb = *(const v2f*)(epack + (cc * NT + n0 + mc) * 2); // global b64
      acc = wmma4(a, bb, acc);
    }
    const float e2 = en2[n0 + mc];
    const int   n  = n0 + mc;
#pragma unroll
    for (int r = 0; r < 8; ++r) {
      const float d = e2 - 2.0f * acc[r];
      if (d < bestv[r]) { bestv[r] = d; besti[r] = n; }   // strict < keeps first
    }
  }
#pragma unroll
  for (int r = 0; r < 8; ++r) {
    const int mr = r + 8 * g;   // D layout: VGPR r, half-wave g -> row r+8g
    rv[wave][mr][mc] = bestv[r];
    ri[wave][mr][mc] = besti[r];
  }
  __syncthreads();

  if (tid < 16) {   // final reduction per row: lexicographic (value, index)
    float bv = 3.4e38f; int bi = 0x7fffffff;
    for (int w = 0; w < 4; ++w)
      for (int c2 = 0; c2 < 16; ++c2) {
        const float v = rv[w][tid][c2]; const int ii = ri[w][tid][c2];
        if (v < bv || (v == bv && ii < bi)) { bv = v; bi = ii; }
      }
    idxs[tid] = bi;
    atomicAdd(&counts[b * NT + bi], 1u);
  }
  __syncthreads();

  // write quantized tensor + accumulate commitment diff
  float fsum = 0.f;
  for (int t = tid; t < 2048; t += 128) {
    const int cc = t >> 4, m = t & 15;
    const int id = idxs[m];
    const float x0 = As[t * 2 + 0], x1 = As[t * 2 + 1];
    const float e0 = embed[(2 * cc) * NT + id];
    const float e1 = embed[(2 * cc + 1) * NT + id];
    q[gbase0 + (2 * cc) * HW + m]     = e0;
    q[gbase0 + (2 * cc + 1) * HW + m] = e1;
    const float d0 = e0 - x0, d1 = e1 - x1;
    fsum += d0 * d0 + d1 * d1;
  }
  red[tid] = fsum; __syncthreads();
  for (int st = 64; st > 0; st >>= 1) {
    if (tid < st) red[tid] += red[tid + st];
    __syncthreads();
  }
  if (tid == 0) wsdiff[s] = red[0];
}

// ---------------------------------------------------------------------------
// K2: 3x3 conv (relu(q), 256 -> 64) + bias + relu, as 9 shifted K=256 GEMMs.
// One 55 KB LDS staging of the relu'd 3x18x256 halo tile shared by 4 waves
// (one per 16-oc tile -> covers all 64 bottleneck channels). 576 WMMAs/wave,
// each fed by exactly 1 ds_load_b64 + 1 global_load_b64 (packed weights).
// ---------------------------------------------------------------------------
__global__ __launch_bounds__(128) void k_res_conv3(const float* __restrict__ q,
                                                   const float* __restrict__ wp,
                                                   const float* __restrict__ b1,
                                                   float* __restrict__ hbuf) {
  __shared__ float Ah[13824];  // pairs at [((cc*3+rr)*18+ww)*2], 55296 B
  const int s = blockIdx.x;
  DECODE_TILE(s, b, hh, w0)
  const int tid = threadIdx.x;

  for (int t = tid; t < 6912; t += 128) {
    const int cc = t / 54, r2 = t % 54, rr = r2 / 18, ww = r2 % 18;
    const int hy = hh + rr - 1, wx = w0 + ww - 1;
    float v0 = 0.f, v1 = 0.f;
    if ((unsigned)hy < 64u && (unsigned)wx < 64u) {
      const long gb = (long)b * CHW + hy * W_ + wx;
      v0 = q[gb + (2 * cc) * HW];     v0 = v0 > 0.f ? v0 : 0.f;   // fused relu
      v1 = q[gb + (2 * cc + 1) * HW]; v1 = v1 > 0.f ? v1 : 0.f;
    }
    Ah[t * 2 + 0] = v0; Ah[t * 2 + 1] = v1;
  }
  __syncthreads();

  const int wave = tid >> 5, lane = tid & 31, g = lane >> 4, mc = lane & 15;
  const int oc = wave * 16 + mc;
  v8f acc = {0.f, 0.f, 0.f, 0.f, 0.f, 0.f, 0.f, 0.f};
  for (int tap = 0; tap < 9; ++tap) {
    const int dh = tap / 3, dw = tap % 3;
    const float* wt = wp + (size_t)tap * WP_PER_TAP * 2;  // [cc][oc] float2
#pragma unroll 8
    for (int k0 = 0; k0 < C_; k0 += 4) {
      const int cc = (k0 >> 1) + g;
      v2f a  = *(const v2f*)(Ah + ((cc * 3 + dh) * 18 + mc + dw) * 2);
      v2f bb = *(const v2f*)(wt + (cc * CB + oc) * 2);   // global_load_b64
      acc = wmma4(a, bb, acc);
    }
  }
  const float bias = b1[oc];
  const long ob = (long)b * (CB * HW) + oc * HW + hh * W_ + w0;
#pragma unroll
  for (int r = 0; r < 8; ++r) {
    const float v = acc[r] + bias;
    hbuf[ob + r + 8 * g] = v > 0.f ? v : 0.f;   // relu before conv2, fused
  }
}

// ---------------------------------------------------------------------------
// K3: 1x1 conv (h, 64 -> 256) + bias + residual add into q. 16 waves share a
// 4 KB staged tile; weight pairs are contiguous -> single b64 B fragments.
// ---------------------------------------------------------------------------
__global__ __launch_bounds__(512) void k_res_conv1(float* __restrict__ q,
                                                   const float* __restrict__ hbuf,
                                                   const float* __restrict__ w2,
                                                   const float* __restrict__ b2) {
  __shared__ float Hs[1024];
  const int s = blockIdx.x;
  DECODE_TILE(s, b, hh, w0)
  const int tid = threadIdx.x;
  {
    const int cc = tid >> 4, m = tid & 15;   // 512 threads = 512 pair slots
    const long gb = (long)b * (CB * HW) + hh * W_ + w0 + m;
    Hs[tid * 2 + 0] = hbuf[gb + (2 * cc) * HW];
    Hs[tid * 2 + 1] = hbuf[gb + (2 * cc + 1) * HW];
  }
  __syncthreads();

  const int wave = tid >> 5, lane = tid & 31, g = lane >> 4, mc = lane & 15;
  const int oc = wave * 16 + mc;
  v8f acc = {0.f, 0.f, 0.f, 0.f, 0.f, 0.f, 0.f, 0.f};
#pragma unroll
  for (int k0 = 0; k0 < CB; k0 += 4) {
    const int cc = (k0 >> 1) + g;
    v2f a  = *(const v2f*)(Hs + (cc * 16 + mc) * 2);
    v2f bb = *(const v2f*)(w2 + oc * CB + k0 + 2 * g);  // contiguous, 8B aligned
    acc = wmma4(a, bb, acc);
  }
  const float bias = b2[oc];
  const long ob = (long)b * CHW + oc * HW + hh * W_ + w0;
#pragma unroll
  for (int r = 0; r < 8; ++r) q[ob + r + 8 * g] += acc[r] + bias;  // residual
}

// ---------------------------------------------------------------------------
// K4: out 1x1 conv (q, 256 -> 3). N-tile zero-padded to 16 in LDS so the WMMA
// loop stays divergence-free (EXEC all-ones); only n<3 columns are stored.
// ---------------------------------------------------------------------------
__global__ __launch_bounds__(32) void k_out_conv(const float* __restrict__ q,
                                                 const float* __restrict__ ow,
                                                 const float* __restrict__ ob,
                                                 float* __restrict__ out) {
  __shared__ float Qs[4096];
  __shared__ float Bs[4096];
  const int s = blockIdx.x;
  DECODE_TILE(s, b, hh, w0)
  const int lane = threadIdx.x, g = lane >> 4, mc = lane & 15;
  const long gbase0 = (long)b * CHW + hh * W_ + w0;

  for (int t = lane; t < 2048; t += 32) {
    const int cc = t >> 4, m = t & 15;     // m doubles as column index for Bs
    Qs[t * 2 + 0] = q[gbase0 + (2 * cc) * HW + m];
    Qs[t * 2 + 1] = q[gbase0 + (2 * cc + 1) * HW + m];
    Bs[t * 2 + 0] = (m < OUTC) ? ow[m * C_ + 2 * cc]     : 0.f;
    Bs[t * 2 + 1] = (m < OUTC) ? ow[m * C_ + 2 * cc + 1] : 0.f;
  }
  __syncthreads();

  v8f acc = {0.f, 0.f, 0.f, 0.f, 0.f, 0.f, 0.f, 0.f};
#pragma unroll 8
  for (int k0 = 0; k0 < C_; k0 += 4) {
    const int cc = (k0 >> 1) + g;
    v2f a  = *(const v2f*)(Qs + (cc * 16 + mc) * 2);
    v2f bb = *(const v2f*)(Bs + (cc * 16 + mc) * 2);
    acc = wmma4(a, bb, acc);
  }
  if (mc < OUTC) {
    const float bias = ob[mc];
    const long obase = ((long)(b * OUTC + mc) * H_ + hh) * W_ + w0;
#pragma unroll
    for (int r = 0; r < 8; ++r) out[obase + r + 8 * g] = acc[r] + bias;
  }
}

// ---------------------------------------------------------------------------
// K5: finalize wordfreq (counts/4096) and diff (tree-reduce 4096 partials).
// Writes every element of the non-quant output region -> no pre-zero needed.
// ---------------------------------------------------------------------------
__global__ __launch_bounds__(256) void k_final(const unsigned* __restrict__ counts,
                                               const float* __restrict__ wsdiff,
                                               float* __restrict__ out) {
  __shared__ float red[256];
  const int gid = blockIdx.x * 256 + threadIdx.x;
  if (blockIdx.x < 32) {
    out[WF_POS + gid] = (float)counts[gid] * (1.0f / 4096.0f);
  } else {
    float sum = 0.f;
    for (int t = threadIdx.x; t < NTILE; t += 256) sum += wsdiff[t];
    red[threadIdx.x] = sum; __syncthreads();
    for (int st = 128; st > 0; st >>= 1) {
      if (threadIdx.x < st) red[threadIdx.x] += red[threadIdx.x + st];
      __syncthreads();
    }
    if (threadIdx.x == 0)
      out[DIFF_POS] = red[0] * (1.0f / ((float)NPOS * (float)C_));
  }
}

// ---------------------------------------------------------------------------
extern "C" void kernel_launch(void* const* d_in, const int* in_sizes, int n_in,
                              void* d_out, int out_size, void* d_ws, size_t ws_size,
                              hipStream_t stream) {
  (void)in_sizes; (void)n_in; (void)out_size; (void)ws_size;
  const float* in  = (const float*)d_in[0];
  const float* emb = (const float*)d_in[1];
  const float* rw1 = (const float*)d_in[2];
  const float* rb1 = (const float*)d_in[3];
  const float* rw2 = (const float*)d_in[4];
  const float* rb2 = (const float*)d_in[5];
  const float* ow  = (const float*)d_in[6];
  const float* ob  = (const float*)d_in[7];
  float* out = (float*)d_out;
  float* ws  = (float*)d_ws;

  float*    q      = ws + Q_OFF;
  float*    hbuf   = ws + H_OFF;
  float*    epack  = ws + EPACK_OFF;
  float*    en2    = ws + EN2_OFF;
  unsigned* counts = (unsigned*)(ws + CNT_OFF);
  float*    wsdiff = ws + WDIFF_OFF;
  float*    wpack  = ws + WPACK_OFF;

  k_prep<<<256, 256, 0, stream>>>(emb, epack, en2, counts);
  k_packw<<<(3 * WP_PER_BLK + 255) / 256, 256, 0, stream>>>(rw1, wpack);
  k_vq<<<NTILE, 128, 0, stream>>>(in, emb, epack, en2, q, counts, wsdiff);
  for (int i = 0; i < 3; ++i) {
    k_res_conv3<<<NTILE, 128, 0, stream>>>(q, wpack + (size_t)i * WP_PER_BLK * 2,
                                           rb1 + i * CB, hbuf);
    k_res_conv1<<<NTILE, 512, 0, stream>>>(q, hbuf, rw2 + (size_t)i * C_ * CB,
                                           rb2 + i * C_);
  }
  k_out_conv<<<NTILE, 32, 0, stream>>>(q, ow, ob, out);
  k_final<<<33, 256, 0, stream>>>(counts, wsdiff, out);
}